// LIFSpike_20091857011333
// MI455X (gfx1250) — compile-verified
//
#include <hip/hip_runtime.h>

// LIF spiking neuron forward pass, x: [B, T, C, H, W] fp32 -> spikes same shape.
//   mem = mem * TAU + x_t ; spike = (mem > THRESH) ; mem = spike ? 0 : mem
// Purely HBM-bound: 268 MB total traffic -> ~11.5 us floor at 23.3 TB/s.

typedef float v4f __attribute__((ext_vector_type(4)));

#define LIF_TAU    0.5f
#define LIF_THRESH 1.0f

static constexpr int T    = 8;
static constexpr int B    = 32;
static constexpr int CHW  = 128 * 32 * 32;  // 131072 (C*H*W, contiguous innermost axes)
static constexpr int CHW4 = CHW / 4;        // 32768 = 2^15 float4 vectors per (b, t) slab

__global__ __launch_bounds__(256) void lif_spike_kernel(const v4f* __restrict__ x,
                                                        v4f* __restrict__ out) {
  // gid in [0, B*CHW4). Each thread owns 4 consecutive floats of one (b, chw)
  // slice across all 8 timesteps.
  const int gid = blockIdx.x * blockDim.x + threadIdx.x;
  const int b   = gid >> 15;            // gid / CHW4
  const int i   = gid & (CHW4 - 1);     // gid % CHW4
  const size_t base = (size_t)b * (size_t)(T * CHW4) + (size_t)i;

  // Issue all 8 timestep loads up front: 8 outstanding global_load_b128 per
  // thread (T-stride = 512 KB, addresses data-independent). NT hint: the
  // 268 MB stream has zero reuse and exceeds the 192 MB L2.
  v4f v[T];
#pragma unroll
  for (int t = 0; t < T; ++t)
    v[t] = __builtin_nontemporal_load(x + base + (size_t)t * CHW4);

  // Serial membrane recurrence, entirely in VGPRs.
  v4f mem = {0.0f, 0.0f, 0.0f, 0.0f};
#pragma unroll
  for (int t = 0; t < T; ++t) {
    mem = mem * LIF_TAU + v[t];         // leak + integrate (v_fma per lane-elem)
    v4f spk;
#pragma unroll
    for (int c = 0; c < 4; ++c) {
      const bool s = mem[c] > LIF_THRESH;  // v_cmp_gt_f32
      spk[c] = s ? 1.0f : 0.0f;            // v_cndmask
      mem[c] = s ? 0.0f : mem[c];          // hard reset (v_cndmask)
    }
    __builtin_nontemporal_store(spk, out + base + (size_t)t * CHW4);  // b128 NT store
  }
}

extern "C" void kernel_launch(void* const* d_in, const int* in_sizes, int n_in,
                              void* d_out, int out_size, void* d_ws, size_t ws_size,
                              hipStream_t stream) {
  (void)in_sizes; (void)n_in; (void)out_size; (void)d_ws; (void)ws_size;
  const v4f* x  = (const v4f*)d_in[0];
  v4f* out      = (v4f*)d_out;

  const int n4    = B * CHW4;   // 1,048,576 vector-threads, covers tensor exactly
  const int block = 256;        // 8 wave32 waves per block
  const int grid  = n4 / block; // 4096 blocks
  lif_spike_kernel<<<grid, block, 0, stream>>>(x, out);
}